// RGATLayer_82360292868670
// MI455X (gfx1250) — compile-verified
//
#include <hip/hip_runtime.h>

// Problem constants (match reference)
#define NN   100000
#define IN_F 128
#define HH   4
#define DD   32
#define HD   128      // H*D
#define RR   3
#define EE   800000
#define NEG_SLOPE 0.2f

typedef __attribute__((ext_vector_type(2))) float v2f;
typedef __attribute__((ext_vector_type(8))) float v8f;

__device__ __forceinline__ float leaky(float x) { return x > 0.f ? x : NEG_SLOPE * x; }

// Order-preserving encode of float into uint so atomicMax(u32) == float max.
__device__ __forceinline__ unsigned encf(float f) {
    unsigned u = __float_as_uint(f);
    return (u & 0x80000000u) ? ~u : (u | 0x80000000u);
}
__device__ __forceinline__ float decf(unsigned u) {
    unsigned b = (u & 0x80000000u) ? (u ^ 0x80000000u) : ~u;
    return __uint_as_float(b);
}

// ---------------------------------------------------------------------------
// Fused GEMM: [N x 128] @ ([128 x 128]src | [128 x 128]dst) + bias -> fs, fd
// One wave computes a 16(M) x 64(N) tile with V_WMMA_F32_16X16X4_F32,
// 4 accumulators reuse one A fragment. 25000 waves total (exact cover).
// A 16x4 f32 layout: lanes0-15 K=0,1 ; lanes16-31 K=2,3 (per ISA table).
// ---------------------------------------------------------------------------
__global__ __launch_bounds__(256) void rgat_gemm_wmma(
    const float* __restrict__ h,
    const float* __restrict__ Wsrc, const float* __restrict__ bsrc,
    const float* __restrict__ Wdst, const float* __restrict__ bdst,
    float* __restrict__ fs, float* __restrict__ fd)
{
    const int wave = (blockIdx.x * blockDim.x + threadIdx.x) >> 5;
    const int lane = threadIdx.x & 31;
    const int half = lane >> 4;
    const int l16  = lane & 15;
    const int tn   = wave & 3;       // 64-col group: 0,1 -> src ; 2,3 -> dst
    const int tm   = wave >> 2;      // 16-row tile: 0..6249 (100000/16 exact)

    const bool  isSrc = (tn < 2);
    const float* __restrict__ W    = isSrc ? Wsrc : Wdst;
    const float* __restrict__ bvec = isSrc ? bsrc : bdst;
    float* __restrict__ ob         = isSrc ? fs   : fd;
    const int cw = (tn & 1) * 64;    // base column within this 128-wide W

    const int row = tm * 16 + l16;
    v8f acc0 = {}; v8f acc1 = {}; v8f acc2 = {}; v8f acc3 = {};

    for (int k = 0; k < IN_F; k += 4) {
        float2 av = *(const float2*)(h + (size_t)row * IN_F + k + 2 * half);
        v2f a; a.x = av.x; a.y = av.y;
        const float* wr0 = W + (size_t)(k + 2 * half) * HD + cw + l16;
        const float* wr1 = wr0 + HD;
        v2f b0; b0.x = wr0[0];  b0.y = wr1[0];
        v2f b1; b1.x = wr0[16]; b1.y = wr1[16];
        v2f b2; b2.x = wr0[32]; b2.y = wr1[32];
        v2f b3; b3.x = wr0[48]; b3.y = wr1[48];
        acc0 = __builtin_amdgcn_wmma_f32_16x16x4_f32(false, a, false, b0, (short)0, acc0, false, false);
        acc1 = __builtin_amdgcn_wmma_f32_16x16x4_f32(false, a, false, b1, (short)0, acc1, false, false);
        acc2 = __builtin_amdgcn_wmma_f32_16x16x4_f32(false, a, false, b2, (short)0, acc2, false, false);
        acc3 = __builtin_amdgcn_wmma_f32_16x16x4_f32(false, a, false, b3, (short)0, acc3, false, false);
    }

    // C/D layout: VGPR v -> M = v (lanes 0-15) / v+8 (lanes 16-31), N = l16.
    const int mbase = tm * 16 + 8 * half;
    const float bb0 = bvec[cw + l16];
    const float bb1 = bvec[cw + 16 + l16];
    const float bb2 = bvec[cw + 32 + l16];
    const float bb3 = bvec[cw + 48 + l16];
#pragma unroll
    for (int v = 0; v < 8; ++v) {
        float* o = ob + (size_t)(mbase + v) * HD + cw + l16;
        o[0]  = acc0[v] + bb0;
        o[16] = acc1[v] + bb1;
        o[32] = acc2[v] + bb2;
        o[48] = acc3[v] + bb3;
    }
}

// ---------------------------------------------------------------------------
// Edge pass 1: logits[e,h] = sum_d leaky(fs[src]+fd[dst]) * attn[h,d]
// plus segment max into mEnc (sortable-uint atomicMax). One wave32 per edge,
// each lane owns 4 contiguous features (lanes 8h..8h+7 belong to head h).
// ---------------------------------------------------------------------------
__global__ __launch_bounds__(256) void rgat_edge_logits(
    const float* __restrict__ fs, const float* __restrict__ fd,
    const int* __restrict__ src, const int* __restrict__ dst,
    const float* __restrict__ attn, float* __restrict__ logits,
    unsigned* __restrict__ mEnc)
{
    const int e    = (blockIdx.x * blockDim.x + threadIdx.x) >> 5;
    const int lane = threadIdx.x & 31;
    if (e >= EE) return;
    const int s = src[e], d = dst[e];

    const float4 fsv = *(const float4*)(fs   + (size_t)s * HD + lane * 4);
    const float4 fdv = *(const float4*)(fd   + (size_t)d * HD + lane * 4);
    const float4 av  = *(const float4*)(attn + lane * 4);

    float p = leaky(fsv.x + fdv.x) * av.x + leaky(fsv.y + fdv.y) * av.y
            + leaky(fsv.z + fdv.z) * av.z + leaky(fsv.w + fdv.w) * av.w;
    // reduce the 8 lanes of each head (wave32)
    p += __shfl_xor(p, 1);
    p += __shfl_xor(p, 2);
    p += __shfl_xor(p, 4);

    const int head = lane >> 3;
    if ((lane & 7) == 0) {
        logits[(size_t)e * HH + head] = p;
        atomicMax(&mEnc[(size_t)d * HH + head], encf(p));
    }
}

// ---------------------------------------------------------------------------
// Edge pass 2: ex = exp(logit - m[dst]); denom[dst] += ex (in place in logits)
// ---------------------------------------------------------------------------
__global__ __launch_bounds__(256) void rgat_edge_exp(
    float* __restrict__ logits, const int* __restrict__ dst,
    const unsigned* __restrict__ mEnc, float* __restrict__ denom)
{
    const int i = blockIdx.x * blockDim.x + threadIdx.x;
    if (i >= EE * HH) return;
    const int e = i >> 2, hh = i & 3;
    const int d = dst[e];
    const float m  = decf(mEnc[(size_t)d * HH + hh]);
    const float ex = __expf(logits[i] - m);
    logits[i] = ex;
    atomicAdd(&denom[(size_t)d * HH + hh], ex);
}

// ---------------------------------------------------------------------------
// Edge pass 3: out[dst] += fs[src] * (ex / denom[dst]). One wave32 per edge,
// 4 fp32 atomics per lane -> hits in the 192MB L2 (out is 51 MB).
// ---------------------------------------------------------------------------
__global__ __launch_bounds__(256) void rgat_edge_scatter(
    const float* __restrict__ fs, const float* __restrict__ logits,
    const int* __restrict__ src, const int* __restrict__ dst,
    const float* __restrict__ denom, float* __restrict__ out)
{
    const int e    = (blockIdx.x * blockDim.x + threadIdx.x) >> 5;
    const int lane = threadIdx.x & 31;
    if (e >= EE) return;
    const int s = src[e], d = dst[e];
    const int head = lane >> 3;
    const float alpha = logits[(size_t)e * HH + head] / denom[(size_t)d * HH + head];
    const float4 fsv = *(const float4*)(fs + (size_t)s * HD + lane * 4);
    float* o = out + (size_t)d * HD + lane * 4;
    atomicAdd(o + 0, fsv.x * alpha);
    atomicAdd(o + 1, fsv.y * alpha);
    atomicAdd(o + 2, fsv.z * alpha);
    atomicAdd(o + 3, fsv.w * alpha);
}

// out[n, c] = sum_r bias[r, c]
__global__ __launch_bounds__(256) void rgat_init_out(
    float* __restrict__ out, const float* __restrict__ bias)
{
    const int i = blockIdx.x * blockDim.x + threadIdx.x;
    if (i >= NN * HD) return;
    const int c = i & (HD - 1);
    out[i] = bias[c] + bias[HD + c] + bias[2 * HD + c];
}

__global__ __launch_bounds__(256) void rgat_init_md(
    unsigned* __restrict__ mEnc, float* __restrict__ denom)
{
    const int i = blockIdx.x * blockDim.x + threadIdx.x;
    if (i >= NN * HH) return;
    mEnc[i]  = 0u;   // encodes "below -inf" for the sortable-uint max
    denom[i] = 0.f;
}

// ---------------------------------------------------------------------------
extern "C" void kernel_launch(void* const* d_in, const int* in_sizes, int n_in,
                              void* d_out, int out_size, void* d_ws, size_t ws_size,
                              hipStream_t stream)
{
    (void)in_sizes; (void)n_in; (void)out_size; (void)ws_size;
    const float* h      = (const float*)d_in[0];
    const float* W_src  = (const float*)d_in[1];
    const float* b_src  = (const float*)d_in[2];
    const float* W_dst  = (const float*)d_in[3];
    const float* b_dst  = (const float*)d_in[4];
    const float* attn   = (const float*)d_in[5];
    const float* bias   = (const float*)d_in[6];
    const int*   src_idx = (const int*)d_in[7];
    const int*   dst_idx = (const int*)d_in[8];
    float* out = (float*)d_out;

    // Workspace layout (reused across relations; stream order serializes):
    //   fs: N*128 f32 (51.2MB) | fd: N*128 f32 | logits: E*H f32 (12.8MB)
    //   mEnc: N*H u32 (1.6MB)  | denom: N*H f32 (1.6MB)   -> ~118.4 MB
    float*    fs     = (float*)d_ws;
    float*    fd     = fs + (size_t)NN * HD;
    float*    logits = fd + (size_t)NN * HD;
    unsigned* mEnc   = (unsigned*)(logits + (size_t)EE * HH);
    float*    denom  = (float*)(mEnc + (size_t)NN * HH);

    rgat_init_out<<<(NN * HD) / 256, 256, 0, stream>>>(out, bias);

    for (int r = 0; r < RR; ++r) {
        const float* Ws = W_src + (size_t)r * IN_F * HD;
        const float* Wd = W_dst + (size_t)r * IN_F * HD;
        const float* bs = b_src + (size_t)r * HD;
        const float* bd = b_dst + (size_t)r * HD;
        const float* at = attn  + (size_t)r * HD;
        const int*   sr = src_idx + (size_t)r * EE;
        const int*   ds = dst_idx + (size_t)r * EE;

        rgat_init_md<<<(NN * HH + 255) / 256, 256, 0, stream>>>(mEnc, denom);
        // 25000 waves (6250 M-tiles x 4 64-col groups), 8 waves/block
        rgat_gemm_wmma<<<3125, 256, 0, stream>>>(h, Ws, bs, Wd, bd, fs, fd);
        // one wave per edge: 800000 / 8 = 100000 blocks
        rgat_edge_logits<<<EE / 8, 256, 0, stream>>>(fs, fd, sr, ds, at, logits, mEnc);
        rgat_edge_exp<<<(EE * HH) / 256, 256, 0, stream>>>(logits, ds, mEnc, denom);
        rgat_edge_scatter<<<EE / 8, 256, 0, stream>>>(fs, logits, sr, ds, denom, out);
    }
}